// Attention_51067161150211
// MI455X (gfx1250) — compile-verified
//
#include <hip/hip_runtime.h>

typedef __attribute__((ext_vector_type(2))) float v2f;
typedef __attribute__((ext_vector_type(8))) float v8f;

#define TT 65    // tweet length
#define QQ 20    // question length
#define DD 100   // feature dim
#define TP 80    // T padded to 5 WMMA tiles
#define QP 32    // Q padded to 2 WMMA tiles
#define DSTR 102 // LDS row stride (even -> b64-mergeable pairs; 38 mod 64 -> conflict-free cols)

__device__ __forceinline__ int iminc(int a, int b) { return a < b ? a : b; }

__launch_bounds__(256)
__global__ void bidaf_attn(const float* __restrict__ H,
                           const float* __restrict__ U,
                           const float* __restrict__ Ws1,
                           const float* __restrict__ Ws2,
                           const float* __restrict__ Ws3,
                           float* __restrict__ G)
{
    __shared__ float sH[TP][DSTR];   // H[b]; rows 65..79 zeroed
    __shared__ float sU[QP][DSTR];   // U[b]; rows 20..31 zeroed
    __shared__ float sS[TP][QP];     // S then (in-place) softmax(at); pad rows stay 0
    __shared__ float sH1[TT];
    __shared__ float sU1[QP];
    __shared__ float sBeta[TT];
    __shared__ float sB[TT];         // softmax over beta
    __shared__ float sHtil[DD];

    const int b    = blockIdx.x;
    const int tid  = threadIdx.x;
    const int lane = tid & 31;
    const int wave = tid >> 5;

    const float* Hb = H + (size_t)b * (TT * DD);
    const float* Ub = U + (size_t)b * (QQ * DD);

    // ---- stage H, U into LDS (vectorized) + zero-fill padding rows ----
    {
        const float4* H4 = reinterpret_cast<const float4*>(Hb);
        for (int i = tid; i < (TT * DD) / 4; i += 256) {
            float4 v = H4[i];
            int f = i * 4;
            int t = f / DD, d = f - t * DD;
            sH[t][d] = v.x; sH[t][d + 1] = v.y; sH[t][d + 2] = v.z; sH[t][d + 3] = v.w;
        }
        const float4* U4 = reinterpret_cast<const float4*>(Ub);
        for (int i = tid; i < (QQ * DD) / 4; i += 256) {
            float4 v = U4[i];
            int f = i * 4;
            int q = f / DD, d = f - q * DD;
            sU[q][d] = v.x; sU[q][d + 1] = v.y; sU[q][d + 2] = v.z; sU[q][d + 3] = v.w;
        }
        for (int i = tid; i < (TP - TT) * DD; i += 256) {
            int t = TT + i / DD, d = i - (i / DD) * DD;
            sH[t][d] = 0.f;
        }
        for (int i = tid; i < (QP - QQ) * DD; i += 256) {
            int q = QQ + i / DD, d = i - (i / DD) * DD;
            sU[q][d] = 0.f;
        }
    }
    __syncthreads();

    // ---- H1[t] = sum_d H*Ws1 ; U1[q] = sum_d U*Ws2 (one wave per row) ----
    for (int t = wave; t < TT; t += 8) {
        float acc = 0.f;
        for (int d = lane; d < DD; d += 32)
            acc += sH[t][d] * Ws1[t * DD + d];
        for (int off = 16; off > 0; off >>= 1)
            acc += __shfl_xor(acc, off, 32);
        if (lane == 0) sH1[t] = acc;
    }
    for (int q = wave; q < QP; q += 8) {
        const int qc = iminc(q, QQ - 1);          // sU pad rows are 0 -> product 0
        float acc = 0.f;
        for (int d = lane; d < DD; d += 32)
            acc += sU[q][d] * Ws2[qc * DD + d];
        for (int off = 16; off > 0; off >>= 1)
            acc += __shfl_xor(acc, off, 32);
        if (lane == 0) sU1[q] = acc;
    }

    // ---- S[t][q] (HU part) = (H .* Ws3) x U^T via V_WMMA_F32_16X16X4_F32 ----
    // All fragment loads unconditional: sH/sU padding rows hold zeros.
    {
        const int m  = lane & 15;
        const int kh = (lane >> 4) << 1;           // 0 (lanes 0-15) or 2 (lanes 16-31)
        const int rowoff = (lane >> 4) << 3;
        for (int tile = wave; tile < 10; tile += 8) {   // 5 M-tiles x 2 N-tiles
            const int mt = tile >> 1, nt = tile & 1;
            const int t0 = mt * 16, q0 = nt * 16;
            const int tA = t0 + m;                 // up to 79: sH row valid (zero if >=65)
            const int qB = q0 + m;                 // up to 31: sU row valid (zero if >=20)
            const float* w3r = Ws3 + iminc(tA, TT - 1) * DD;  // clamp; zero sH row kills it
            v8f c = {};
            #pragma unroll
            for (int k0 = 0; k0 < DD; k0 += 4) {   // 25 exact K-steps
                const int d0 = k0 + kh;
                v2f a, bb;
                a.x  = sH[tA][d0]     * w3r[d0];
                a.y  = sH[tA][d0 + 1] * w3r[d0 + 1];
                bb.x = sU[qB][d0];
                bb.y = sU[qB][d0 + 1];
                c = __builtin_amdgcn_wmma_f32_16x16x4_f32(false, a, false, bb,
                                                          (short)0, c, false, false);
            }
            #pragma unroll
            for (int j = 0; j < 8; ++j)
                sS[t0 + rowoff + j][q0 + m] = c[j];
        }
    }
    __syncthreads();

    // ---- per-row softmax over q (with H1+U1 added); beta[t] = row max ----
    for (int t = wave; t < TT; t += 8) {
        float s = (lane < QQ) ? (sS[t][lane] + sH1[t] + sU1[lane]) : -__builtin_inff();
        float mx = s;
        for (int off = 16; off > 0; off >>= 1)
            mx = fmaxf(mx, __shfl_xor(mx, off, 32));
        float e = (lane < QQ) ? __expf(s - mx) : 0.f;
        float sum = e;
        for (int off = 16; off > 0; off >>= 1)
            sum += __shfl_xor(sum, off, 32);
        sS[t][lane] = e / sum;                     // lanes >= 20 store 0
        if (lane == 0) sBeta[t] = mx;
    }
    __syncthreads();

    // ---- softmax over beta (length 65) on wave 0 ----
    if (wave == 0) {
        float v0 = (lane < TT)      ? sBeta[lane]      : -__builtin_inff();
        float v1 = (lane + 32 < TT) ? sBeta[lane + 32] : -__builtin_inff();
        float v2 = (lane + 64 < TT) ? sBeta[lane + 64] : -__builtin_inff();
        float mx = fmaxf(v0, fmaxf(v1, v2));
        for (int off = 16; off > 0; off >>= 1)
            mx = fmaxf(mx, __shfl_xor(mx, off, 32));
        float e0 = (lane < TT)      ? __expf(v0 - mx) : 0.f;
        float e1 = (lane + 32 < TT) ? __expf(v1 - mx) : 0.f;
        float e2 = (lane + 64 < TT) ? __expf(v2 - mx) : 0.f;
        float sum = e0 + e1 + e2;
        for (int off = 16; off > 0; off >>= 1)
            sum += __shfl_xor(sum, off, 32);
        float inv = 1.f / sum;
        if (lane < TT)      sB[lane]      = e0 * inv;
        if (lane + 32 < TT) sB[lane + 32] = e1 * inv;
        if (lane + 64 < TT) sB[lane + 64] = e2 * inv;
    }
    __syncthreads();

    // ---- Htil[d] = sum_t b[t] * H[t][d] ----
    if (tid < DD) {
        float acc = 0.f;
        for (int t = 0; t < TT; ++t)
            acc += sB[t] * sH[t][tid];
        sHtil[tid] = acc;
    }
    __syncthreads();

    // ---- Util = at x U via WMMA; fuse output of G from C registers ----
    // sS rows 65..79 are exact zeros; columns >= DD are garbage but discarded on store.
    {
        const int m  = lane & 15;
        const int kh = (lane >> 4) << 1;
        const int rowoff = (lane >> 4) << 3;
        for (int tile = wave; tile < 35; tile += 8) {   // 5 M-tiles x 7 N-tiles
            const int mt = tile / 7, nt = tile % 7;
            const int t0 = mt * 16, d0 = nt * 16;
            const int tA = t0 + m;                 // up to 79: sS row valid
            const int dcol = d0 + m;               // up to 111
            const int dc = iminc(dcol, DD - 1);    // clamp; bad cols discarded below
            v8f c = {};
            #pragma unroll
            for (int k0 = 0; k0 < QQ; k0 += 4) {   // 5 exact K-steps over q
                const int q0 = k0 + kh;
                v2f a, bb;
                a.x  = sS[tA][q0];
                a.y  = sS[tA][q0 + 1];
                bb.x = sU[q0][dc];
                bb.y = sU[q0 + 1][dc];
                c = __builtin_amdgcn_wmma_f32_16x16x4_f32(false, a, false, bb,
                                                          (short)0, c, false, false);
            }
            if (dcol < DD) {
                const float htil = sHtil[dcol];
                #pragma unroll
                for (int j = 0; j < 8; ++j) {
                    const int t = t0 + rowoff + j;
                    if (t < TT) {
                        const float h = sH[t][dcol];
                        const float u = c[j];
                        size_t base = ((size_t)b * TT + t) * (4 * DD);
                        G[base + dcol]          = h;        // H
                        G[base + DD + dcol]     = u;        // Util
                        G[base + 2 * DD + dcol] = h * u;    // H*Util
                        G[base + 3 * DD + dcol] = h * htil; // H*Htil
                    }
                }
            }
        }
    }
}

extern "C" void kernel_launch(void* const* d_in, const int* in_sizes, int n_in,
                              void* d_out, int out_size, void* d_ws, size_t ws_size,
                              hipStream_t stream) {
    (void)n_in; (void)out_size; (void)d_ws; (void)ws_size;
    const float* H   = (const float*)d_in[0];
    const float* U   = (const float*)d_in[1];
    const float* Ws1 = (const float*)d_in[2];
    const float* Ws2 = (const float*)d_in[3];
    const float* Ws3 = (const float*)d_in[4];
    float* G = (float*)d_out;
    const int B = in_sizes[0] / (TT * DD);   // 8192
    bidaf_attn<<<dim3(B), dim3(256), 0, stream>>>(H, U, Ws1, Ws2, Ws3, G);
}